// BatchHighOrderActivation_83502754168911
// MI455X (gfx1250) — compile-verified
//
#include <hip/hip_runtime.h>

typedef float v8f __attribute__((ext_vector_type(8)));
typedef float v2f __attribute__((ext_vector_type(2)));

#define ARITY   4
#define IN_DIM  512
#define OUT_DIM 32
#define BATCH   4096
#define NVERT   16          // 2^ARITY simplex vertices

// Block = 256 threads = 8 waves. Each wave: 16 batches x 1 channel.
// Grid = (BATCH/16, IN_DIM/8).
__global__ __launch_bounds__(256)
void BatchHighOrderActivation_wmma_kernel(const float* __restrict__ X,
                                          const float* __restrict__ P,
                                          float* __restrict__ out)
{
    // 8 channels' param tables: 8 * 16 * 32 floats = 16 KB
    __shared__ __align__(16) float sP[8 * NVERT * OUT_DIM];
    // per-wave scattered coefficient matrices: 8 waves * 16 batches * 16 verts = 8 KB
    __shared__ __align__(16) float sC[8 * 16 * NVERT];

    const int tid  = threadIdx.x;
    const int wave = tid >> 5;
    const int lane = tid & 31;

    const int batchBase = blockIdx.x * 16;
    const int gBase     = blockIdx.y * 8;
    const int g         = gBase + wave;

    // ---- Stage params for 8 consecutive channels: 16 KB contiguous ----
    {
        const float4* s4 = (const float4*)(P + (size_t)gBase * (NVERT * OUT_DIM));
        float4*       d4 = (float4*)sP;
#pragma unroll
        for (int i = 0; i < 4; ++i)
            d4[tid + i * 256] = s4[tid + i * 256];
    }

    // ---- Build scattered coefficient matrix Coef[16 batches][16 verts] in LDS ----
    float* myC = sC + wave * (16 * NVERT);
    {
        float4 z = {0.f, 0.f, 0.f, 0.f};
        ((float4*)myC)[lane]      = z;   // zero 256 floats per wave
        ((float4*)myC)[lane + 32] = z;
    }
    if (lane < 16) {
        const int b = batchBase + lane;
        const float4 xv = *(const float4*)(X + ((size_t)b * IN_DIM + g) * ARITY);
        float v0 = xv.x, v1 = xv.y, v2 = xv.z, v3 = xv.w;
        int   i0 = 0,    i1 = 1,    i2 = 2,    i3 = 3;
        // 5-comparator sorting network, ascending (ties irrelevant to the result)
#define CSW(va, vb, ia, ib)                         \
        {   bool sw = (va) > (vb);                  \
            float fmx = sw ? (va) : (vb);           \
            float fmn = sw ? (vb) : (va);           \
            int   imx = sw ? (ia) : (ib);           \
            int   imn = sw ? (ib) : (ia);           \
            (va) = fmn; (vb) = fmx; (ia) = imn; (ib) = imx; }
        CSW(v0, v1, i0, i1);
        CSW(v2, v3, i2, i3);
        CSW(v0, v2, i0, i2);
        CSW(v1, v3, i1, i3);
        CSW(v1, v2, i1, i2);
#undef CSW
        // gap coefficients
        const float c0 = v0;
        const float c1 = v1 - v0;
        const float c2 = v2 - v1;
        const float c3 = v3 - v2;
        // reverse-cumsum bitmask indices (always 4 distinct slots, idx0 == 15)
        const int idx3 = 1 << i3;
        const int idx2 = idx3 + (1 << i2);
        const int idx1 = idx2 + (1 << i1);
        float* row = myC + lane * NVERT;
        row[15]   = c0;
        row[idx1] = c1;
        row[idx2] = c2;
        row[idx3] = c3;
    }
    __syncthreads();

    // ---- D[16x32] = Coef[16x16] x P[16x32] via V_WMMA_F32_16X16X4_F32 ----
    // 32-bit A (16x4) layout: lanes 0-15 hold M=lane, K={kb,kb+1}; lanes 16-31 hold K={kb+2,kb+3}.
    // B (4x16) mirrors it across N: lanes 0-15 = N, VGPR0/1 = K rows {0,1} (lo half) / {2,3} (hi half).
    const float* pg    = sP + wave * (NVERT * OUT_DIM);
    const int    row   = lane & 15;
    const int    khalf = (lane >> 4) << 1;     // 0 or 2

    v8f acc0 = {};
    v8f acc1 = {};
#pragma unroll
    for (int kk = 0; kk < 4; ++kk) {
        const int kb = kk * 4 + khalf;
        v2f a;
        a.x = myC[row * NVERT + kb];
        a.y = myC[row * NVERT + kb + 1];
        v2f b0, b1;
        b0.x = pg[(kb)     * OUT_DIM + row];        // N-tile 0 (cols 0..15)
        b0.y = pg[(kb + 1) * OUT_DIM + row];
        b1.x = pg[(kb)     * OUT_DIM + 16 + row];   // N-tile 1 (cols 16..31)
        b1.y = pg[(kb + 1) * OUT_DIM + 16 + row];
        // (neg_a, A, neg_b, B, c_mod, C, reuse_a, reuse_b)
        acc0 = __builtin_amdgcn_wmma_f32_16x16x4_f32(false, a, false, b0,
                                                     (short)0, acc0, false, false);
        acc1 = __builtin_amdgcn_wmma_f32_16x16x4_f32(false, a, false, b1,
                                                     (short)0, acc1, false, false);
    }

    // ---- Store: C/D layout: VGPR v -> (M = v + 8*(lane/16), N = lane%16) ----
    const int n    = lane & 15;
    const int mOff = (lane >> 4) << 3;              // 0 or 8
    float* obase = out + (size_t)batchBase * (IN_DIM * OUT_DIM) + (size_t)g * OUT_DIM;
#pragma unroll
    for (int v = 0; v < 8; ++v) {
        const size_t roff = (size_t)(v + mOff) * (IN_DIM * OUT_DIM);
        __builtin_nontemporal_store(acc0[v], obase + roff + n);
        __builtin_nontemporal_store(acc1[v], obase + roff + 16 + n);
    }
}

extern "C" void kernel_launch(void* const* d_in, const int* in_sizes, int n_in,
                              void* d_out, int out_size, void* d_ws, size_t ws_size,
                              hipStream_t stream) {
    const float* X = (const float*)d_in[0];   // [BATCH, IN_DIM, ARITY] f32
    const float* P = (const float*)d_in[1];   // [IN_DIM, 16, OUT_DIM]  f32
    float* out = (float*)d_out;               // [BATCH, IN_DIM, OUT_DIM] f32

    dim3 grid(BATCH / 16, IN_DIM / 8);
    BatchHighOrderActivation_wmma_kernel<<<grid, 256, 0, stream>>>(X, P, out);
}